// MNIST_CNN_PLLay_CnnTakesDtm_63496796504407
// MI455X (gfx1250) — compile-verified
//
#include <hip/hip_runtime.h>
#include <hip/hip_bf16.h>
#include <math.h>

typedef __attribute__((ext_vector_type(2))) float v2f;
typedef __attribute__((ext_vector_type(8))) float v8f;

#define GRID_N 28
#define P_N    784
#define B_N    16
#define BY2    (1.0f / (13.5f * 13.5f))
#define MAXD2  1459   // max (di^2+dj^2) = 2*27^2 = 1458 -> buckets [0,1458]

// ---------------------------------------------------------------------------
// Build ORDER (stable distance-sorted neighbor indices) and D2S per pixel.
// Counting sort over integer squared distances; stable by index (tie order
// does not affect _dtm's clipped budget sum, so any stable order is valid).
// ---------------------------------------------------------------------------
__global__ void build_order_kernel(int* __restrict__ order, float* __restrict__ d2s) {
    __shared__ int cnt[MAXD2];
    const int p  = blockIdx.x;
    const int pi = p / GRID_N, pj = p % GRID_N;
    for (int i = threadIdx.x; i < MAXD2; i += blockDim.x) cnt[i] = 0;
    __syncthreads();
    for (int q = threadIdx.x; q < P_N; q += blockDim.x) {
        int qi = q / GRID_N, qj = q % GRID_N;
        int di = pi - qi, dj = pj - qj;
        atomicAdd(&cnt[di * di + dj * dj], 1);
    }
    __syncthreads();
    if (threadIdx.x == 0) {
        int run = 0;
        for (int i = 0; i < MAXD2; ++i) { int c = cnt[i]; cnt[i] = run; run += c; }
        for (int q = 0; q < P_N; ++q) {
            int qi = q / GRID_N, qj = q % GRID_N;
            int di = pi - qi, dj = pj - qj;
            int d2 = di * di + dj * dj;
            int pos = cnt[d2]++;
            order[p * P_N + pos] = q;
            d2s  [p * P_N + pos] = (float)d2 * BY2;
        }
    }
}

// ---------------------------------------------------------------------------
// 3x3 SAME conv + bias + relu, NHWC in / HWIO weights.
// out[b*outStride + outOffset + pix*Cout + co]
// ---------------------------------------------------------------------------
__global__ void conv3x3_relu_kernel(const float* __restrict__ in, const float* __restrict__ w,
                                    const float* __restrict__ bias, float* __restrict__ out,
                                    int Cin, int Cout, int outStride, int outOffset) {
    int idx = blockIdx.x * blockDim.x + threadIdx.x;
    int total = B_N * P_N * Cout;
    if (idx >= total) return;
    int co  = idx % Cout;
    int pix = (idx / Cout) % P_N;
    int b   = idx / (Cout * P_N);
    int y = pix / GRID_N, x = pix % GRID_N;
    float acc = bias[co];
    for (int ky = 0; ky < 3; ++ky) {
        int yy = y + ky - 1;
        if (yy < 0 || yy >= GRID_N) continue;
        for (int kx = 0; kx < 3; ++kx) {
            int xx = x + kx - 1;
            if (xx < 0 || xx >= GRID_N) continue;
            const float* ip = in + ((size_t)b * P_N + (size_t)yy * GRID_N + xx) * Cin;
            const float* wp = w + (size_t)((ky * 3 + kx) * Cin) * Cout + co;
            for (int ci = 0; ci < Cin; ++ci)
                acc += ip[ci] * wp[(size_t)ci * Cout];
        }
    }
    out[(size_t)b * outStride + outOffset + (size_t)pix * Cout + co] = fmaxf(acc, 0.0f);
}

// ---------------------------------------------------------------------------
// Distance-to-measure transform for both m0 values in one pass.
// One block per (b, 256-pixel chunk); x row cached in LDS.
// ---------------------------------------------------------------------------
__global__ void dtm_kernel(const float* __restrict__ x, const int* __restrict__ order,
                           const float* __restrict__ d2s,
                           float* __restrict__ out1, float* __restrict__ out2) {
    __shared__ float sx[P_N];
    __shared__ float sred[256];
    const int b = blockIdx.y;
    for (int i = threadIdx.x; i < P_N; i += blockDim.x) sx[i] = x[(size_t)b * P_N + i];
    __syncthreads();
    float ps = 0.0f;
    for (int i = threadIdx.x; i < P_N; i += blockDim.x) ps += sx[i];
    sred[threadIdx.x] = ps;
    __syncthreads();
    for (int s = 128; s > 0; s >>= 1) {
        if (threadIdx.x < (unsigned)s) sred[threadIdx.x] += sred[threadIdx.x + s];
        __syncthreads();
    }
    const float sumw = sred[0];
    const int p = blockIdx.x * blockDim.x + threadIdx.x;
    if (p >= P_N) return;
    const float m0W1 = 0.05f * sumw, m0W2 = 0.2f * sumw;
    float cum = 0.0f, acc1 = 0.0f, acc2 = 0.0f;
    const int*   op = order + (size_t)p * P_N;
    const float* dp = d2s   + (size_t)p * P_N;
    for (int j = 0; j < P_N; ++j) {
        if ((j & 63) == 0) {
            __builtin_prefetch(op + j + 128, 0, 1);   // global_prefetch_b8
            __builtin_prefetch(dp + j + 128, 0, 1);
        }
        float wv = sx[op[j]];
        float d2 = dp[j];
        float r1 = fminf(fmaxf(m0W1 - cum, 0.0f), wv);
        float r2 = fminf(fmaxf(m0W2 - cum, 0.0f), wv);
        acc1 += r1 * d2;
        acc2 += r2 * d2;
        cum  += wv;
    }
    out1[(size_t)b * P_N + p] = sqrtf(acc1 / m0W1);
    out2[(size_t)b * P_N + p] = sqrtf(acc2 / m0W2);
}

// ---------------------------------------------------------------------------
// Persistence landscape: tent values over P, top-K per (b, t) -> lam(B, T*K).
// One wave32 per (b, t).
// ---------------------------------------------------------------------------
__global__ void topo_lam_kernel(const float* __restrict__ f, float* __restrict__ lam,
                                float t0, float dt, float tmax, int T, int K) {
    __shared__ float stop[32 * 3];
    const int bt = blockIdx.x;
    const int b = bt / T, t = bt % T;
    const float tv  = t0 + dt * (float)t;
    const float cap = tmax - tv;
    const float* fp = f + (size_t)b * P_N;
    float v0 = 0.0f, v1 = 0.0f, v2 = 0.0f;
    for (int p = threadIdx.x; p < P_N; p += 32) {
        float tent = fmaxf(fminf(tv - fp[p], cap), 0.0f);
        if (tent > v0)      { v2 = v1; v1 = v0; v0 = tent; }
        else if (tent > v1) { v2 = v1; v1 = tent; }
        else if (tent > v2) { v2 = tent; }
    }
    stop[threadIdx.x * 3 + 0] = v0;
    stop[threadIdx.x * 3 + 1] = v1;
    stop[threadIdx.x * 3 + 2] = v2;
    __syncthreads();
    if (threadIdx.x == 0) {
        float w0 = 0.0f, w1 = 0.0f, w2 = 0.0f;
        for (int i = 0; i < 96; ++i) {
            float v = stop[i];
            if (v > w0)      { w2 = w1; w1 = w0; w0 = v; }
            else if (v > w1) { w2 = w1; w1 = v; }
            else if (v > w2) { w2 = v; }
        }
        float* o = lam + (size_t)b * (T * K) + (size_t)t * K;
        o[0] = w0; o[1] = w1;
        if (K == 3) o[2] = w2;
    }
}

// ---------------------------------------------------------------------------
// One-wave 16x16 output tile of A(16xK) @ B(KxN-tile), f32 WMMA (16x16x4).
// A layout per lane: lanes 0-15 -> M=lane, K={k0,k0+1}; lanes 16-31 -> K={k0+2,k0+3}.
// B mirrors with N over lanes. D: vgpr r, lane l, half h -> D[r+8h][l].
// ---------------------------------------------------------------------------
__device__ __forceinline__ v8f wmma_tile_f32(const float* A, int lda,
                                             const float* Bm, int ldb,
                                             int K, int lane) {
    v8f acc = {};
    const int h = lane >> 4, l = lane & 15;
    const int K4 = K & ~3;
    for (int k0 = 0; k0 < K4; k0 += 4) {
        int ka = k0 + 2 * h;
        v2f a, bv;
        a.x  = A[(size_t)l * lda + ka];
        a.y  = A[(size_t)l * lda + ka + 1];
        bv.x = Bm[(size_t)ka * ldb + l];
        bv.y = Bm[(size_t)(ka + 1) * ldb + l];
        acc = __builtin_amdgcn_wmma_f32_16x16x4_f32(false, a, false, bv,
                                                    (short)0, acc, false, false);
    }
    for (int k = K4; k < K; ++k) {          // scalar remainder (K % 4)
        float bkl = Bm[(size_t)k * ldb + l];
#pragma unroll
        for (int r = 0; r < 8; ++r)
            acc[r] += A[(size_t)(r + 8 * h) * lda + k] * bkl;
    }
    return acc;
}

// out(16x16) = relu(A(16xK) @ W(Kx16) + bias), written at column offset.
__global__ void gemm16_relu_kernel(const float* __restrict__ A, const float* __restrict__ W,
                                   const float* __restrict__ bias, float* __restrict__ out,
                                   int K, int outStride, int outOffset) {
    const int lane = threadIdx.x & 31;
    v8f acc = wmma_tile_f32(A, K, W, 16, K, lane);
    const int h = lane >> 4, l = lane & 15;
#pragma unroll
    for (int r = 0; r < 8; ++r) {
        int m = r + 8 * h;
        out[(size_t)m * outStride + outOffset + l] = fmaxf(acc[r] + bias[l], 0.0f);
    }
}

// Fused head: h = relu(xc(16x848) @ W7(848x64) + b7); out = h @ W8(64x10) + b8.
// 4 waves tile N=64; wave 0 does the second GEMM against zero-padded LDS W8.
__global__ void final_fc_kernel(const float* __restrict__ xc, const float* __restrict__ W7,
                                const float* __restrict__ b7, const float* __restrict__ W8,
                                const float* __restrict__ b8, float* __restrict__ out) {
    __shared__ float hbuf[16 * 64];
    __shared__ float w8p[64 * 16];
    const int tid = threadIdx.x;
    for (int i = tid; i < 64 * 16; i += 128) {
        int k = i >> 4, n = i & 15;
        w8p[i] = (n < 10) ? W8[(size_t)k * 10 + n] : 0.0f;
    }
    const int wave = tid >> 5, lane = tid & 31;
    const int h = lane >> 4, l = lane & 15;
    const int n0 = wave * 16;
    v8f acc = wmma_tile_f32(xc, 848, W7 + n0, 64, 848, lane);
#pragma unroll
    for (int r = 0; r < 8; ++r) {
        int m = r + 8 * h;
        int n = n0 + l;
        hbuf[m * 64 + n] = fmaxf(acc[r] + b7[n], 0.0f);
    }
    __syncthreads();
    if (wave == 0) {
        v8f acc2 = wmma_tile_f32(hbuf, 64, w8p, 16, 64, lane);
        if (l < 10) {
#pragma unroll
            for (int r = 0; r < 8; ++r) {
                int m = r + 8 * h;
                out[(size_t)m * 10 + l] = acc2[r] + b8[l];
            }
        }
    }
}

// ---------------------------------------------------------------------------
extern "C" void kernel_launch(void* const* d_in, const int* in_sizes, int n_in,
                              void* d_out, int out_size, void* d_ws, size_t ws_size,
                              hipStream_t stream) {
    (void)in_sizes; (void)n_in; (void)out_size; (void)ws_size;
    const float* x    = (const float*)d_in[0];
    const float* k11  = (const float*)d_in[1];
    const float* cb11 = (const float*)d_in[2];
    const float* k12  = (const float*)d_in[3];
    const float* cb12 = (const float*)d_in[4];
    const float* k31  = (const float*)d_in[5];
    const float* cb31 = (const float*)d_in[6];
    const float* k32  = (const float*)d_in[7];
    const float* cb32 = (const float*)d_in[8];
    const float* k41  = (const float*)d_in[9];
    const float* cb41 = (const float*)d_in[10];
    const float* k42  = (const float*)d_in[11];
    const float* cb42 = (const float*)d_in[12];
    const float* W33  = (const float*)d_in[13];
    const float* b33  = (const float*)d_in[14];
    const float* W43  = (const float*)d_in[15];
    const float* b43  = (const float*)d_in[16];
    const float* W5   = (const float*)d_in[17];
    const float* b5   = (const float*)d_in[18];
    const float* W6   = (const float*)d_in[19];
    const float* b6   = (const float*)d_in[20];
    const float* W7   = (const float*)d_in[21];
    const float* b7   = (const float*)d_in[22];
    const float* W8   = (const float*)d_in[23];
    const float* b8   = (const float*)d_in[24];
    float* out = (float*)d_out;

    char* ws = (char*)d_ws;
    size_t off = 0;
    auto carve = [&](size_t bytes) -> char* {
        char* p = ws + off;
        off += (bytes + 255) & ~(size_t)255;
        return p;
    };
    int*   order  = (int*)  carve((size_t)P_N * P_N * sizeof(int));
    float* d2s    = (float*)carve((size_t)P_N * P_N * sizeof(float));
    float* hidden = (float*)carve((size_t)B_N * P_N * 32 * sizeof(float));
    float* x21    = (float*)carve((size_t)B_N * P_N * sizeof(float));
    float* x22    = (float*)carve((size_t)B_N * P_N * sizeof(float));
    float* f3     = (float*)carve((size_t)B_N * P_N * sizeof(float));
    float* f4     = (float*)carve((size_t)B_N * P_N * sizeof(float));
    float* lamA   = (float*)carve((size_t)B_N * 50 * sizeof(float));
    float* lamB   = (float*)carve((size_t)B_N * 81 * sizeof(float));
    float* lamC   = (float*)carve((size_t)B_N * 50 * sizeof(float));
    float* lamD   = (float*)carve((size_t)B_N * 81 * sizeof(float));
    float* xc     = (float*)carve((size_t)B_N * 848 * sizeof(float));

    const float tA0 = 0.06f, tAd = 0.24f / 24.0f, tAm = 0.3f;  // TSEQ_A
    const float tB0 = 0.14f, tBd = 0.26f / 26.0f, tBm = 0.4f;  // TSEQ_B
    const int cg32 = (B_N * P_N * 32 + 255) / 256;
    const int cg1  = (B_N * P_N + 255) / 256;

    build_order_kernel<<<P_N, 256, 0, stream>>>(order, d2s);

    // Path 1: x -> conv(k11) -> conv(k12) -> x1 (into xc cols [0,784))
    conv3x3_relu_kernel<<<cg32, 256, 0, stream>>>(x, k11, cb11, hidden, 1, 32, P_N * 32, 0);
    conv3x3_relu_kernel<<<cg1,  256, 0, stream>>>(hidden, k12, cb12, xc, 32, 1, 848, 0);

    // DTM (both m0 in one pass)
    dtm_kernel<<<dim3(4, B_N), 256, 0, stream>>>(x, order, d2s, x21, x22);

    // Path 3: x2_1 -> conv(k31) -> conv(k32) -> f3
    conv3x3_relu_kernel<<<cg32, 256, 0, stream>>>(x21, k31, cb31, hidden, 1, 32, P_N * 32, 0);
    conv3x3_relu_kernel<<<cg1,  256, 0, stream>>>(hidden, k32, cb32, f3, 32, 1, P_N, 0);

    // Path 4: x2_2 -> conv(k41) -> conv(k42) -> f4
    conv3x3_relu_kernel<<<cg32, 256, 0, stream>>>(x22, k41, cb41, hidden, 1, 32, P_N * 32, 0);
    conv3x3_relu_kernel<<<cg1,  256, 0, stream>>>(hidden, k42, cb42, f4, 32, 1, P_N, 0);

    // Landscape features
    topo_lam_kernel<<<B_N * 25, 32, 0, stream>>>(f3,  lamA, tA0, tAd, tAm, 25, 2);
    topo_lam_kernel<<<B_N * 27, 32, 0, stream>>>(f4,  lamB, tB0, tBd, tBm, 27, 3);
    topo_lam_kernel<<<B_N * 25, 32, 0, stream>>>(x21, lamC, tA0, tAd, tAm, 25, 2);
    topo_lam_kernel<<<B_N * 27, 32, 0, stream>>>(x22, lamD, tB0, tBd, tBm, 27, 3);

    // WMMA GEMMs: x3/x4/x5/x6 into xc columns 784/800/816/832
    gemm16_relu_kernel<<<1, 32, 0, stream>>>(lamA, W33, b33, xc, 50, 848, 784);
    gemm16_relu_kernel<<<1, 32, 0, stream>>>(lamB, W43, b43, xc, 81, 848, 800);
    gemm16_relu_kernel<<<1, 32, 0, stream>>>(lamC, W5,  b5,  xc, 50, 848, 816);
    gemm16_relu_kernel<<<1, 32, 0, stream>>>(lamD, W6,  b6,  xc, 81, 848, 832);

    // Fused head (WMMA x2)
    final_fc_kernel<<<1, 128, 0, stream>>>(xc, W7, b7, W8, b8, out);
}